// DrugAttentionLayer_16810501996742
// MI455X (gfx1250) — compile-verified
//
#include <hip/hip_runtime.h>
#include <hip/hip_bf16.h>
#include <math.h>

#define ALPHA 0.2f
#define DIM 128

typedef __attribute__((ext_vector_type(2))) float v2f;
typedef __attribute__((ext_vector_type(8))) float v8f;

// ---------------------------------------------------------------------------
// Kernel 1: S = E @ [a1 a2] using V_WMMA_F32_16X16X4_F32 (exact f32).
// One wave handles 16 rows of E. Per K-step of 4:
//   A (16x4):  lanes 0-15 hold K=k,k+1 for M=lane; lanes 16-31 hold K=k+2,k+3.
//   B (4x16):  lane = N column; only N=0 (a1) and N=1 (a2) nonzero.
//              VGPR0/1 hold K=half*2, half*2+1 (mirror of A striping).
//   C/D 16x16: VGPR r: lanes 0-15 -> M=r, lanes 16-31 -> M=r+8; N=lane&15.
// si[m] = D[m][0] (lanes 0/16), sj[m] = D[m][1] (lanes 1/17).
// ---------------------------------------------------------------------------
__global__ void gemv_si_sj(const float* __restrict__ emb,
                           const float* __restrict__ aphim,
                           float* __restrict__ si, float* __restrict__ sj,
                           int N) {
  const int wave  = (int)((blockIdx.x * blockDim.x + threadIdx.x) >> 5);
  const int lane  = threadIdx.x & 31;
  const int tiles = (N + 15) >> 4;
  if (wave >= tiles) return;                 // wave-uniform branch (EXEC stays all-1)

  const int m0   = wave << 4;
  const int half = lane >> 4;                // 0: K lo pair, 1: K hi pair
  const int col  = lane & 15;                // A: row-in-tile, B: N column
  const int rowm = m0 + col;
  const int rowc = rowm < N ? rowm : N - 1;  // clamp (dup loads, stores guarded)
  const float bmask = (col < 2) ? 1.0f : 0.0f;
  const float* av   = aphim + (col < 2 ? col : 0) * DIM;  // a1 or a2 column
  const float* erow = emb + (long)rowc * DIM + half * 2;

  v8f c = {};
  #pragma unroll
  for (int k = 0; k < DIM; k += 4) {
    v2f a = *(const v2f*)(erow + k);         // 8B-aligned b64 load
    const int koff = k + half * 2;
    v2f b;
    b.x = av[koff]     * bmask;
    b.y = av[koff + 1] * bmask;
    // (neg_a, A, neg_b, B, c_mod, C, reuse_a, reuse_b)
    c = __builtin_amdgcn_wmma_f32_16x16x4_f32(false, a, false, b,
                                              (short)0, c, false, false);
  }

  if (col < 2) {
    float* dst = (col == 0) ? si : sj;
    const int mbase = m0 + half * 8;
    #pragma unroll
    for (int r = 0; r < 8; ++r) {
      const int m = mbase + r;
      if (m < N) dst[m] = c[r];
    }
  }
}

// ---- float <-> monotone unsigned mapping (for atomic segment-max) ----------
__device__ __forceinline__ unsigned fmap(float x) {
  unsigned u = __float_as_uint(x);
  return (u & 0x80000000u) ? ~u : (u | 0x80000000u);
}
__device__ __forceinline__ float funmap(unsigned u) {
  return __uint_as_float((u & 0x80000000u) ? (u ^ 0x80000000u) : ~u);
}

__global__ void init_nodes(unsigned* __restrict__ mmap,
                           float* __restrict__ denom, int N) {
  const int i = blockIdx.x * blockDim.x + threadIdx.x;
  if (i < N) { mmap[i] = 0u; denom[i] = 0.0f; }  // 0u < fmap(any finite e)
}

// Kernel 2: e = leaky_relu(si[ctr]+sj[nbr]); segment-max via uint atomicMax.
__global__ void edge_logits(const float* __restrict__ si,
                            const float* __restrict__ sj,
                            const int* __restrict__ rel,
                            float* __restrict__ ews,
                            unsigned* __restrict__ mmap, int E) {
  const int i = blockIdx.x * blockDim.x + threadIdx.x;
  if (i >= 2 * E) return;
  int c, nb;
  if (i < E) { c = rel[2 * i];     nb = rel[2 * i + 1]; }
  else       { const int j = i - E; c = rel[2 * j + 1]; nb = rel[2 * j]; }
  const float v = si[c] + sj[nb];
  const float e = v > 0.0f ? v : ALPHA * v;
  ews[i] = e;
  atomicMax(&mmap[c], fmap(e));
}

// Kernel 3: ex = exp(e - m[ctr]); denom[ctr] += ex  (ex overwrites e in ws).
__global__ void edge_exp(const int* __restrict__ rel,
                         float* __restrict__ ews,
                         const unsigned* __restrict__ mmap,
                         float* __restrict__ denom, int E) {
  const int i = blockIdx.x * blockDim.x + threadIdx.x;
  if (i >= 2 * E) return;
  const int c = (i < E) ? rel[2 * i] : rel[2 * (i - E) + 1];
  const float m  = funmap(mmap[c]);
  const float ex = __expf(ews[i] - m);
  ews[i] = ex;
  atomicAdd(&denom[c], ex);
}

// Kernel 4: out[ctr] += (ex/denom[ctr]) * emb[nbr].  One wave per edge
// instance; each lane owns 4 of 128 dims (float4 gather + 4 f32 atomics).
__global__ void edge_agg(const int* __restrict__ rel,
                         const float* __restrict__ ews,
                         const float* __restrict__ denom,
                         const float* __restrict__ emb,
                         float* __restrict__ out, int E) {
  const long gid  = (long)blockIdx.x * blockDim.x + threadIdx.x;
  const long inst = gid >> 5;
  const int  lane = (int)(gid & 31);
  if (inst >= 2L * E) return;
  int c, nb;
  if (inst < E) { c = rel[2 * inst];  nb = rel[2 * inst + 1]; }
  else { const long j = inst - E; c = rel[2 * j + 1]; nb = rel[2 * j]; }
  const float w = ews[inst] / denom[c];
  const float4 v = *(const float4*)(emb + (long)nb * DIM + lane * 4);
  float* o = out + (long)c * DIM + lane * 4;
  atomicAdd(o + 0, w * v.x);
  atomicAdd(o + 1, w * v.y);
  atomicAdd(o + 2, w * v.z);
  atomicAdd(o + 3, w * v.w);
}

extern "C" void kernel_launch(void* const* d_in, const int* in_sizes, int n_in,
                              void* d_out, int out_size, void* d_ws, size_t ws_size,
                              hipStream_t stream) {
  const float* emb   = (const float*)d_in[0];
  const int*   rel   = (const int*)d_in[1];   // jax x64 off -> int32 pairs
  const float* aphim = (const float*)d_in[2]; // 256 floats: a1 | a2

  const int N = in_sizes[0] / DIM;
  const int E = in_sizes[1] / 2;
  float* out = (float*)d_out;

  // Workspace layout: si[N] | sj[N] | mmap[N] (uint) | denom[N] | ews[2E]
  float*    si    = (float*)d_ws;
  float*    sj    = si + N;
  unsigned* mmap  = (unsigned*)(sj + N);
  float*    denom = (float*)(mmap + N);
  float*    ews   = denom + N;

  // out = emb (residual term); graph-capture-safe d2d copy
  hipMemcpyAsync(out, emb, (size_t)N * DIM * sizeof(float),
                 hipMemcpyDeviceToDevice, stream);

  const int tiles = (N + 15) / 16;                 // one wave per 16-row tile
  gemv_si_sj<<<(tiles + 7) / 8, 256, 0, stream>>>(emb, aphim, si, sj, N);

  init_nodes<<<(N + 255) / 256, 256, 0, stream>>>(mmap, denom, N);

  const int EI = 2 * E;
  edge_logits<<<(EI + 255) / 256, 256, 0, stream>>>(si, sj, rel, ews, mmap, E);
  edge_exp   <<<(EI + 255) / 256, 256, 0, stream>>>(rel, ews, mmap, denom, E);

  const long aggThreads = (long)EI * 32;           // one wave per edge instance
  edge_agg<<<(int)((aggThreads + 255) / 256), 256, 0, stream>>>(
      rel, ews, denom, emb, out, E);
}